// FNO_3D_87522843558046
// MI455X (gfx1250) — compile-verified
//
#include <hip/hip_runtime.h>
#include <hip/hip_bf16.h>

// ---------------------------------------------------------------------------
// FNO-3D for MI455X (gfx1250), f32 end-to-end, WMMA f32 16x16x4 matrix pipe.
//
// Truncated rfftn (16 modes/axis) == small dense DFT GEMMs. Pipeline/layer:
//   S1: z-rDFT   (real->cplx)  v[x,y,z,c]        -> F1[x,y,kz,c]   (67 MB)
//   S2: y-DFT    (cplx)        F1                -> F2[x,ky,kz,c]  (8.4 MB)
//   S3: x-DFT    (cplx)        F2                -> F3[kx,ky,kz,c] (1 MB)
//   S4: per-mode complex 32x32 channel mix with Rphi
//   S5: inv-x expand (cplx)    G3 -> G2[x,ky,kz,c]
//   S6: inv-y expand (cplx)    G2 -> G1[x,y,kz,c]
//   S7: fused z-irDFT + v@w + exact GELU -> next v  (async LDS staging)
// Bandwidth-bound (~1.1 GB/layer @ 23.3 TB/s); compute ~60 GFLOP total, so
// f32 WMMA keeps reference precision at negligible cost.
// ---------------------------------------------------------------------------

typedef __attribute__((ext_vector_type(2))) float v2f;
typedef __attribute__((ext_vector_type(4))) float v4f;
typedef __attribute__((ext_vector_type(8))) float v8f;
typedef int v4i __attribute__((vector_size(16)));   // matches builtin param type

static __device__ __forceinline__ v8f wmma4(v2f a, v2f b, v8f c) {
  // D(16x16 f32) = A(16x4 f32) x B(4x16 f32) + C   -> v_wmma_f32_16x16x4_f32
  return __builtin_amdgcn_wmma_f32_16x16x4_f32(
      /*neg_a=*/false, a, /*neg_b=*/false, b,
      /*c_mod=*/(short)0, c, /*reuse_a=*/false, /*reuse_b=*/false);
}

static __device__ __forceinline__ float gelu_exact(float x) {
  return 0.5f * x * (1.0f + erff(x * 0.70710678118654752f));
}

// --- gfx1250 async global->LDS copy (ASYNCcnt path), with safe fallback ----
#if defined(__has_builtin)
#if __has_builtin(__builtin_amdgcn_global_load_async_to_lds_b128) && \
    __has_builtin(__builtin_amdgcn_s_wait_asynccnt)
#define FNO_HAS_ASYNC_LDS 1
#endif
#endif

#if defined(FNO_HAS_ASYNC_LDS)
static __device__ __forceinline__ void async_cp16(void* ldsDst, const void* gSrc) {
  __builtin_amdgcn_global_load_async_to_lds_b128(
      (__attribute__((address_space(1))) v4i*)gSrc,
      (__attribute__((address_space(3))) v4i*)ldsDst, 0, 0);
}
#endif

// ------------------------- table construction ------------------------------
// Interleaved (cos,sin) pair tables so one 16B load yields a full A-fragment.
// Fcs[k][t] = {cos(2pi k t/128), -sin(...)}                       [16x128x2]
// Ecs[x][k] = {cos(2pi k x/128)/128, +sin(...)/128}               [128x16x2]
// Czs[z][k] = {Cz, Sz}: Cz[z][0]=1/128, Cz=2cos/128; Sz=-2sin/128 [128x16x2]
__global__ __launch_bounds__(256) void init_tables(float* Fcs, float* Ecs,
                                                   float* Czs) {
  const double PI2 = 6.283185307179586476925286766559;
  for (int i = threadIdx.x; i < 16 * 128; i += 256) {
    int k = i / 128, t = i % 128;
    double th = PI2 * (double)(k * t) / 128.0;
    Fcs[2 * i] = (float)cos(th);
    Fcs[2 * i + 1] = (float)(-sin(th));
  }
  for (int i = threadIdx.x; i < 128 * 16; i += 256) {
    int x = i / 16, k = i % 16;
    double th = PI2 * (double)(k * x) / 128.0;
    Ecs[2 * i] = (float)(cos(th) / 128.0);
    Ecs[2 * i + 1] = (float)(sin(th) / 128.0);
    Czs[2 * i] = (k == 0) ? (1.0f / 128.0f) : (float)(2.0 * cos(th) / 128.0);
    Czs[2 * i + 1] = (k == 0) ? 0.0f : (float)(-2.0 * sin(th) / 128.0);
  }
}

// ------------------------------- lifting -----------------------------------
__global__ __launch_bounds__(256) void lift_kernel(const float* __restrict__ x,
                                                   const float* __restrict__ win,
                                                   const float* __restrict__ bin,
                                                   float* __restrict__ v) {
  long t = (long)blockIdx.x * 256 + threadIdx.x;   // t = p*32 + c
  long p = t >> 5;
  int c = (int)(t & 31);
  float acc = bin[c];
  const float* xp = x + p * 4;
  #pragma unroll
  for (int d = 0; d < 4; ++d) acc = fmaf(xp[d], win[d * 32 + c], acc);
  v[t] = gelu_exact(acc);
}

// ------------------- generic strided complex DFT GEMM ----------------------
// Out[b, k, j] = sum_t (Dc[k,t] + i*Ds[k,t]) * In[b, t, j]
// One wave computes a 16(k) x 16(j) tile of Re and Im outputs.
// CPLX=false -> real input fast path (stage 1), branch-free specialization.
template <bool CPLX>
__global__ __launch_bounds__(256) void dft_gemm(
    const float* __restrict__ inR, const float* __restrict__ inI,
    float* __restrict__ outR, float* __restrict__ outI,
    const float* __restrict__ Dcs, int dStride,
    int T, int nTilesK, int nTilesJ, int nBatch,
    long inBS, long inTS, long outBS, long outKS) {
  int wid = (int)((blockIdx.x * blockDim.x + threadIdx.x) >> 5);
  int lane = threadIdx.x & 31;
  int jt = wid % nTilesJ;
  int tmp = wid / nTilesJ;
  int kt = tmp % nTilesK;
  int b = tmp / nTilesK;
  if (b >= nBatch) return;                 // wave-uniform exit
  int row = lane & 15;                     // A: M-row / B: N-col
  int h = lane >> 4;                       // K sub-index (2h, 2h+1)

  // A: interleaved (cos,sin) pairs; 16B load covers t = 2h, 2h+1.
  const float* dp = Dcs + 2 * ((long)(kt * 16 + row) * dStride + 2 * h);
  const float* pr0 = inR + (long)b * inBS + jt * 16 + row + (long)(2 * h) * inTS;
  const float* pr1 = pr0 + inTS;
  const float* pi0 = nullptr;
  const float* pi1 = nullptr;
  if (CPLX) {
    pi0 = inI + (long)b * inBS + jt * 16 + row + (long)(2 * h) * inTS;
    pi1 = pi0 + inTS;
  }
  const long step = 4 * inTS;

  v8f accR = {0.f, 0.f, 0.f, 0.f, 0.f, 0.f, 0.f, 0.f};
  v8f accI = {0.f, 0.f, 0.f, 0.f, 0.f, 0.f, 0.f, 0.f};

  for (int t0 = 0; t0 < T; t0 += 4) {
    v4f q = *(const v4f*)dp;               // {c(ta), s(ta), c(ta+1), s(ta+1)}
    v2f ac = {q.x, q.z};
    v2f as = {q.y, q.w};
    v2f br = {*pr0, *pr1};
    accR = wmma4(ac, br, accR);            // Re += Dc*Ir
    accI = wmma4(as, br, accI);            // Im += Ds*Ir
    if (CPLX) {
      v2f bi = {*pi0, *pi1};
      v2f asn = {-q.y, -q.w};
      accR = wmma4(asn, bi, accR);         // Re -= Ds*Ii
      accI = wmma4(ac, bi, accI);          // Im += Dc*Ii
      pi0 += step;
      pi1 += step;
    }
    dp += 8;
    pr0 += step;
    pr1 += step;
  }

  long ob = (long)b * outBS + jt * 16 + row;
  #pragma unroll
  for (int r = 0; r < 8; ++r) {
    int kk = kt * 16 + r + 8 * h;          // C/D layout: M = r + 8h
    outR[ob + (long)kk * outKS] = accR[r];
    outI[ob + (long)kk * outKS] = accI[r];
  }
}

// -------------------- per-mode complex channel mix (S4) --------------------
// G3[m,e] = sum_d (F3r+i F3i)[m,d] * (Rr + i Ri)[m,d,e]   (4096 modes)
__global__ __launch_bounds__(256) void mode_mix(
    const float* __restrict__ F3r, const float* __restrict__ F3i,
    const float* __restrict__ Rr, const float* __restrict__ Ri,
    float* __restrict__ G3r, float* __restrict__ G3i) {
  int m = (int)((blockIdx.x * blockDim.x + threadIdx.x) >> 5);
  int e = threadIdx.x & 31;
  const float* fr = F3r + (long)m * 32;
  const float* fi = F3i + (long)m * 32;
  const float* rr = Rr + (long)m * 1024;
  const float* ri = Ri + (long)m * 1024;
  float ar = 0.f, ai = 0.f;
  #pragma unroll 4
  for (int d = 0; d < 32; ++d) {
    float frd = fr[d], fid = fi[d];
    float rrv = rr[d * 32 + e], riv = ri[d * 32 + e];
    ar = fmaf(frd, rrv, fmaf(-fid, riv, ar));
    ai = fmaf(frd, riv, fmaf(fid, rrv, ai));
  }
  G3r[(long)m * 32 + e] = ar;
  G3i[(long)m * 32 + e] = ai;
}

// --------- fused z-irDFT + pointwise v@w + exact GELU (S7) -----------------
// One block per (x,y) pencil: out[z,c] = gelu(irfft_z(G1)[z,c] + sum_d v[z,d]w[d,c])
// v slab staged via GLOBAL_LOAD_ASYNC_TO_LDS_B128 when available.
// vS row stride = 36 floats (144B): 16B-aligned rows + conflict-free banks.
__global__ __launch_bounds__(256) void finalize_layer(
    const float* __restrict__ G1r, const float* __restrict__ G1i,
    const float* __restrict__ v, const float* __restrict__ w,
    const float* __restrict__ Czs, float* __restrict__ u) {
  __shared__ __align__(16) float vS[128 * 36];
  __shared__ __align__(16) float grS[16 * 32];
  __shared__ __align__(16) float giS[16 * 32];
  __shared__ __align__(16) float wS[32 * 32];

  long xy = blockIdx.x;
  int tid = threadIdx.x;
  const float* vp = v + xy * 4096;
  const float* grp = G1r + xy * 512;
  const float* gip = G1i + xy * 512;

#if defined(FNO_HAS_ASYNC_LDS)
  for (int t = tid; t < 1024; t += 256) {  // v slab: 128 rows x 8 xfers
    int z = t >> 3, kk = t & 7;
    async_cp16(&vS[z * 36 + kk * 4], vp + z * 32 + kk * 4);
  }
  if (tid < 128) {
    async_cp16(&grS[tid * 4], grp + tid * 4);
    async_cp16(&giS[tid * 4], gip + tid * 4);
  }
  async_cp16(&wS[tid * 4], w + tid * 4);   // 256 xfers cover 1024 floats
  __builtin_amdgcn_s_wait_asynccnt(0);
#else
  for (int i = tid; i < 4096; i += 256) vS[(i >> 5) * 36 + (i & 31)] = vp[i];
  for (int i = tid; i < 512; i += 256) { grS[i] = grp[i]; giS[i] = gip[i]; }
  for (int i = tid; i < 1024; i += 256) wS[i] = w[i];
#endif
  __syncthreads();

  int wave = tid >> 5, lane = tid & 31;
  int row = lane & 15, h = lane >> 4;
  float* up = u + xy * 4096;

  for (int tt = wave; tt < 16; tt += 8) {  // 8 z-tiles x 2 c-tiles, wave-uniform
    int zt = tt & 7, ct = tt >> 3;
    v8f acc = {0.f, 0.f, 0.f, 0.f, 0.f, 0.f, 0.f, 0.f};
    int zr = zt * 16 + row;

    // irDFT over kz: acc += Cz * G1r + Sz * G1i   (scales folded into tables)
    #pragma unroll
    for (int k0 = 0; k0 < 16; k0 += 4) {
      int ka = k0 + 2 * h;
      v4f q = *(const v4f*)(Czs + 2 * (zr * 16 + ka));
      v2f a_c = {q.x, q.z};
      v2f a_s = {q.y, q.w};
      v2f b_r = {grS[ka * 32 + ct * 16 + row], grS[(ka + 1) * 32 + ct * 16 + row]};
      v2f b_i = {giS[ka * 32 + ct * 16 + row], giS[(ka + 1) * 32 + ct * 16 + row]};
      acc = wmma4(a_c, b_r, acc);
      acc = wmma4(a_s, b_i, acc);
    }
    // pointwise channel GEMM: acc += v(z,d) * w(d,c)
    #pragma unroll
    for (int d0 = 0; d0 < 32; d0 += 4) {
      int da = d0 + 2 * h;
      v2f a_v = {vS[zr * 36 + da], vS[zr * 36 + da + 1]};
      v2f b_w = {wS[da * 32 + ct * 16 + row], wS[(da + 1) * 32 + ct * 16 + row]};
      acc = wmma4(a_v, b_w, acc);
    }
    #pragma unroll
    for (int r = 0; r < 8; ++r) {
      int z = zt * 16 + r + 8 * h, c = ct * 16 + row;
      up[z * 32 + c] = gelu_exact(acc[r]);
    }
  }
}

// ------------------------------ projection ---------------------------------
__global__ __launch_bounds__(256) void proj_kernel(const float* __restrict__ v,
                                                   const float* __restrict__ wout,
                                                   const float* __restrict__ bout,
                                                   float* __restrict__ out) {
  long p = (long)blockIdx.x * 256 + threadIdx.x;
  const float* vp = v + p * 32;
  float acc = bout[0];
  #pragma unroll
  for (int c = 0; c < 32; ++c) acc = fmaf(vp[c], wout[c], acc);
  out[p] = acc;
}

// ------------------------------- launcher ----------------------------------
extern "C" void kernel_launch(void* const* d_in, const int* in_sizes, int n_in,
                              void* d_out, int out_size, void* d_ws, size_t ws_size,
                              hipStream_t stream) {
  const float* x = (const float*)d_in[0];
  const float* win = (const float*)d_in[1];
  const float* bin = (const float*)d_in[2];
  const float* Rr[4] = {(const float*)d_in[3], (const float*)d_in[6],
                        (const float*)d_in[9], (const float*)d_in[12]};
  const float* Ri[4] = {(const float*)d_in[4], (const float*)d_in[7],
                        (const float*)d_in[10], (const float*)d_in[13]};
  const float* wl[4] = {(const float*)d_in[5], (const float*)d_in[8],
                        (const float*)d_in[11], (const float*)d_in[14]};
  const float* wout = (const float*)d_in[15];
  const float* bout = (const float*)d_in[16];
  float* out = (float*)d_out;

  // ---- workspace carve (floats); total ~615 MB ----
  float* p = (float*)d_ws;
  float* Fcs = p; p += 4096;       // 16x128 (cos,sin) pairs
  float* Ecs = p; p += 4096;       // 128x16 pairs
  float* Czs = p; p += 4096;       // 128x16 pairs
  float* vA = p;  p += 67108864;   // 128^3 x 32
  float* vB = p;  p += 67108864;
  float* F1r = p; p += 8388608;    // 128x128x16x32
  float* F1i = p; p += 8388608;
  float* F2r = p; p += 1048576;    // 128x16x16x32
  float* F2i = p; p += 1048576;
  float* F3r = p; p += 131072;     // 16^3 x 32
  float* F3i = p; p += 131072;
  float* G3r = p; p += 131072;
  float* G3i = p; p += 131072;
  // inverse-path buffers alias forward ones (consumed before overwritten):
  float* G2r = F2r; float* G2i = F2i;
  float* G1r = F1r; float* G1i = F1i;

  init_tables<<<1, 256, 0, stream>>>(Fcs, Ecs, Czs);
  lift_kernel<<<262144, 256, 0, stream>>>(x, win, bin, vA);

  float* cur = vA;
  float* nxt = vB;
  for (int L = 0; L < 4; ++L) {
    // S1: z-rDFT (real input): batch=(x,y)=16384, T=128, j=c(32)
    dft_gemm<false><<<4096, 256, 0, stream>>>(cur, nullptr, F1r, F1i, Fcs, 128,
                                              128, 1, 2, 16384, 4096, 32, 512, 32);
    // S2: y-DFT: batch=x(128), T=128(y), j=(kz,c)=512
    dft_gemm<true><<<512, 256, 0, stream>>>(F1r, F1i, F2r, F2i, Fcs, 128,
                                            128, 1, 32, 128, 65536, 512, 8192, 512);
    // S3: x-DFT: single GEMM, T=128(x), j=(ky,kz,c)=8192
    dft_gemm<true><<<64, 256, 0, stream>>>(F2r, F2i, F3r, F3i, Fcs, 128,
                                           128, 1, 512, 1, 0, 8192, 0, 8192);
    // S4: per-mode complex 32x32 channel mix
    mode_mix<<<512, 256, 0, stream>>>(F3r, F3i, Rr[L], Ri[L], G3r, G3i);
    // S5: inverse expand x: T=16(kx), Kout=128(x), j=8192
    dft_gemm<true><<<512, 256, 0, stream>>>(G3r, G3i, G2r, G2i, Ecs, 16,
                                            16, 8, 512, 1, 0, 8192, 0, 8192);
    // S6: inverse expand y: batch=x(128), T=16(ky), Kout=128(y), j=512
    dft_gemm<true><<<4096, 256, 0, stream>>>(G2r, G2i, G1r, G1i, Ecs, 16,
                                             16, 8, 32, 128, 8192, 512, 65536, 512);
    // S7: fused z-irDFT + v@w + GELU
    finalize_layer<<<16384, 256, 0, stream>>>(G1r, G1i, cur, wl[L], Czs, nxt);
    float* t2 = cur; cur = nxt; nxt = t2;
  }

  proj_kernel<<<8192, 256, 0, stream>>>(cur, wout, bout, out);
}